// BiologicalLIFNeuron_58695023067773
// MI455X (gfx1250) — compile-verified
//
#include <hip/hip_runtime.h>
#include <stdint.h>

// LIF spiking-neuron scan for MI455X (gfx1250, wave32).
// One thread per neuron (b,h); 512-step recurrence held in registers.
// input_embedding staged through LDS, double-buffered, via the CDNA5
// Tensor Data Mover (TENSOR_LOAD_TO_LDS, one 2D-tile DMA per time-tile,
// waited with s_wait_tensorcnt).  Fallback: per-lane
// GLOBAL_LOAD_ASYNC_TO_LDS_B32 + s_wait_asynccnt if the TDM builtin is
// unavailable.  Spike stores are non-temporal (single-touch stream).

#if __has_builtin(__builtin_amdgcn_tensor_load_to_lds)
#define HAVE_TDM 1
#else
#define HAVE_TDM 0
#endif

namespace {

constexpr int Bn  = 8;
constexpr int Sn  = 512;
constexpr int Hn  = 2048;
constexpr int BLK = 256;          // 8 wave32 per workgroup
constexpr int TT  = 8;            // timesteps staged per DMA round (8 KB/block)
constexpr int NT  = Sn / TT;      // 64 tiles

constexpr float DT        = 1.0e-3f;
constexpr float A_MEM     = 0.95122942f;   // exp(-dt/0.02)
constexpr float A_SYN     = 0.81873075f;   // exp(-dt/0.005)
constexpr float ADAPT_DEC = 0.99004983f;   // exp(-dt/0.1)
constexpr float STDP_DEC  = 0.95122942f;   // exp(-dt/0.02)
constexpr float TWO_PI    = 6.28318530718f;

typedef __attribute__((ext_vector_type(4))) unsigned int v4u;
typedef __attribute__((ext_vector_type(4))) int          v4i;
typedef __attribute__((ext_vector_type(8))) int          v8i;

__device__ __forceinline__ float clampf(float x, float lo, float hi) {
  return fminf(fmaxf(x, lo), hi);
}

// Counter-based 32-bit mixer (deterministic: pure function of (id, t, stream)).
__device__ __forceinline__ uint32_t mix3(uint32_t a, uint32_t b, uint32_t c) {
  uint32_t x = a * 0x9E3779B9u + b * 0x85EBCA6Bu + c * 0xC2B2AE35u + 0x27D4EB2Fu;
  x ^= x >> 16; x *= 0x7FEB352Du;
  x ^= x >> 15; x *= 0x846CA68Bu;
  x ^= x >> 16;
  return x;
}

__device__ __forceinline__ float u01(uint32_t h) {
  return (float)(h >> 8) * (1.0f / 16777216.0f);
}

struct F2 { float a, b; };

// Box-Muller pair from two hashed uniforms (fast transcendental path).
__device__ __forceinline__ F2 nrm2(uint32_t id, uint32_t t, uint32_t s) {
  uint32_t h1 = mix3(id, t, 2u * s + 11u);
  uint32_t h2 = mix3(id, t, 2u * s + 12u);
  float u1 = u01(h1) + 1.0e-12f;
  float u2 = u01(h2);
  float r  = sqrtf(-2.0f * __logf(u1));
  float sn, cs;
  __sincosf(TWO_PI * u2, &sn, &cs);
  F2 o; o.a = r * cs; o.b = r * sn; return o;
}

__global__ __launch_bounds__(BLK)
void lif_scan_kernel(const float* __restrict__ inp,          // [B,S,H]
                     const float* __restrict__ p_amv,
                     const float* __restrict__ p_asv,
                     const float* __restrict__ p_mn,
                     const float* __restrict__ p_synN,
                     const float* __restrict__ p_tn,
                     const float* __restrict__ p_pink,
                     const float* __restrict__ p_jit,
                     const float* __restrict__ p_homeo,
                     const float* __restrict__ p_tb,
                     const float* __restrict__ p_astr,
                     const float* __restrict__ p_bp,
                     const float* __restrict__ p_bc,
                     const float* __restrict__ p_phase,
                     const float* __restrict__ p_seed,
                     const float* __restrict__ vth_off,      // [H]
                     const float* __restrict__ nfac_v,       // [H]
                     float* __restrict__ out)                // [B,S,H]
{
  __shared__ float smem[2][TT][BLK];

  const int tid = threadIdx.x;
  const int gi  = blockIdx.x * BLK + tid;    // 0 .. B*H-1
  const int b   = gi >> 11;                  // / H
  const int h   = gi & (Hn - 1);
  const int h0  = (blockIdx.x & 7) * BLK;    // block's h start
  const uint32_t nid = (uint32_t)gi;
  const bool wave0 = (threadIdx.x >> 5) == 0;   // uniform per wave

  // ---- scalar params (uniform; hoisted, clamped as in reference) ----
  const float amv    = clampf(p_amv[0],   0.1f,   0.3f);
  const float asv    = clampf(p_asv[0],   0.1f,   0.25f);
  const float mn     = clampf(p_mn[0],    1.0f,   2.5f);
  const float synN   = clampf(p_synN[0],  0.1f,   1.5f);
  const float tn     = clampf(p_tn[0],    0.0f,   5.0f);
  const float pinkC  = clampf(p_pink[0],  0.5f,   2.0f) * 0.1f;
  const float jitC   = clampf(p_jit[0],   0.2f,   1.2f);
  const float homeo  = clampf(p_homeo[0], 0.5f,   2.0f);
  const float th_b   = p_tb[0];
  const float astr   = clampf(p_astr[0],  0.0f,   0.1f);
  const float bp     = clampf(p_bp[0],    0.001f, 0.01f);
  const float bc     = clampf(p_bc[0],    0.5f,   1.5f);
  const float phase  = p_phase[0];
  const float seedR  = p_seed[0];                       // raw (chaos_mod)
  const float csd    = clampf(seedR, 0.5f, 2.0f);       // clamped (chaosn)
  const float vth_o  = vth_off[h];
  const float nfac   = nfac_v[h];

  // ---- initial state (matches state=None branch) ----
  F2 i0 = nrm2(nid, 0xFFFF0000u, 0);
  F2 i1 = nrm2(nid, 0xFFFF0000u, 1);
  float v     = -65.0f + i0.a * 3.0f;
  float syn   = i0.b * 0.02f;
  float adapt = i1.a * 0.02f;
  float refr  = 0.0f;
  float stdp  = 0.0f;

  const float* gbase  = inp + (size_t)b * Sn * Hn + h;    // per-thread column
  const float* gtile0 = inp + (size_t)b * Sn * Hn + h0;   // block tile origin
  float*       obase  = out + (size_t)b * Sn * Hn + h;
  const uint32_t smem_base = (uint32_t)(uintptr_t)&smem[0][0][0];
  (void)gbase; (void)gtile0; (void)smem_base;

#if HAVE_TDM
  // One TDM descriptor moves the whole 8x256-float strided tile into LDS.
  // D# packing per CDNA5 ISA 8.3/8.4: group0 = {flags, lds_addr, global_addr,
  // type}, group1 = {data_size, tensor dims, tile dims, dim0 stride}.
  auto issue_tdm = [&](int tile, int bufIdx) {
    const uint64_t ga = (uint64_t)(uintptr_t)(gtile0 + (size_t)(tile * TT) * Hn);
    const uint32_t lds = smem_base + (uint32_t)bufIdx * (TT * BLK * 4);
    constexpr uint32_t TD0 = 1u << 20;   // tensor_dim0 (>= tile_dim0; no OOB)
    constexpr uint32_t TD1 = 1u << 20;   // tensor_dim1 (>= tile_dim1; no OOB)
    v4u g0;
    g0[0] = 1u;                                            // count=1, user mode
    g0[1] = lds;                                           // lds_addr (bytes)
    g0[2] = (uint32_t)ga;                                  // global_addr[31:0]
    g0[3] = (uint32_t)((ga >> 32) & 0x1FFFFFFu) | (2u << 30); // addr[56:32]|type=2
    v8i g1;
    g1[0] = (int)(2u << 16);                               // data_size = 4 bytes
    g1[1] = (int)((TD0 & 0xFFFFu) << 16);                  // tensor_dim0 lo
    g1[2] = (int)(((TD0 >> 16) & 0xFFFFu) | ((TD1 & 0xFFFFu) << 16));
    g1[3] = (int)(((TD1 >> 16) & 0xFFFFu) | ((uint32_t)BLK << 16)); // tile_dim0=256
    g1[4] = (int)TT;                                       // tile_dim1=8, dim2=0
    g1[5] = (int)Hn;                                       // tensor_dim0_stride
    g1[6] = 0;
    g1[7] = 0;
    v4i z4 = {0, 0, 0, 0};
#if __clang_major__ >= 23
    v8i z8 = {0, 0, 0, 0, 0, 0, 0, 0};
    __builtin_amdgcn_tensor_load_to_lds(g0, g1, z4, z4, z8, 0);
#else
    __builtin_amdgcn_tensor_load_to_lds(g0, g1, z4, z4, 0);
#endif
  };
#else
  // Fallback: per-lane fire-and-forget async copies (ASYNCcnt-tracked).
  auto issue_async = [&](int tile, int bufIdx) {
    const float* gp = gbase + (size_t)(tile * TT) * Hn;
#pragma unroll
    for (int k = 0; k < TT; ++k) {
      uint32_t lds = (uint32_t)(uintptr_t)&smem[bufIdx][k][tid];
      const float* g = gp + (size_t)k * Hn;
      asm volatile("global_load_async_to_lds_b32 %0, %1, off"
                   :: "v"(lds), "v"(g) : "memory");
    }
  };
#endif

  // ---- prologue: stage tile 0 into buffer 0 ----
#if HAVE_TDM
  if (wave0) {
    issue_tdm(0, 0);
    __builtin_amdgcn_s_wait_tensorcnt(0);
  }
  __syncthreads();
#else
  issue_async(0, 0);
#endif

  for (int tile = 0; tile < NT; ++tile) {
    const int buf = tile & 1;

#if HAVE_TDM
    if (tile + 1 < NT && wave0) issue_tdm(tile + 1, buf ^ 1);
#else
    if (tile + 1 < NT) {
      issue_async(tile + 1, buf ^ 1);
      // newest 8 (next tile) stay in flight; older 8 complete in order.
      asm volatile("s_wait_asynccnt 0x8" ::: "memory");
    } else {
      asm volatile("s_wait_asynccnt 0x0" ::: "memory");
    }
#endif

    for (int k = 0; k < TT; ++k) {
      const int t = tile * TT + k;
      const uint32_t tu = (uint32_t)t;
      const float ct = (float)t * DT;
      const float x  = smem[buf][k][tid];

      // per-step scalar alpha jitter (same stream for every neuron, as in ref)
      F2 gs = nrm2(0x5EEDBEEFu, tu, 0);
      const float a_mem = A_MEM * (1.0f + gs.a * amv);
      const float a_syn = A_SYN * (1.0f + gs.b * asv);

      // ---- fused input-current construction ----
      F2 n0 = nrm2(nid, tu, 1);
      const float baseline = 2.0f * (1.0f + n0.a * 0.1f);
      const float upois = u01(mix3(nid, tu, 101u));
      const float pk = (upois > 0.904837f)
                         ? ((upois > 0.995321f) ? ((upois > 0.999845f) ? 3.0f : 2.0f)
                                                : 1.0f)
                         : 0.0f;                                  // ~Poisson(0.1)
      const float poisson_n = pk * synN * n0.b;
      F2 n1 = nrm2(nid, tu, 2);
      const float bg = n1.a * 0.5f * u01(mix3(nid, tu, 103u));
      F2 n2 = nrm2(nid, tu, 3);
      F2 n3 = nrm2(nid, tu, 4);
      const float pink = (n1.b + 0.5f * n2.a + 0.25f * n2.b + 0.125f * n3.a) * pinkC;
      F2 n4 = nrm2(nid, tu, 5);
      const float jitter = n3.b * jitC * __sinf(n4.a * 10.0f);
      const float theta  = __sinf(TWO_PI * 8.0f  * ct + phase)        * 0.05f;
      const float gam    = __sinf(TWO_PI * 40.0f * ct + 2.0f * phase) * 0.02f;
      const float chaos_mod = __sinf(seedR + (float)t * 0.1f) * n4.b * 0.1f;
      const float I = x * (50.0f * 0.45f) * homeo + 0.02f /* target*homeo*2 */
                    + baseline
                    + (poisson_n + bg + pink + jitter + theta + gam + chaos_mod);

      // ---- LIF state update ----
      const float stdp_mod = 1.0f + 0.5f * tanhf(stdp * 10.0f);
      syn = a_syn * syn + (1.0f - a_syn) * I * stdp_mod;

      F2 m0 = nrm2(nid, tu, 6);
      const float lognorm  = __expf(m0.a * 0.3f) * mn - 1.0f;
      const float indiv    = m0.b * nfac;
      F2 m1 = nrm2(nid, tu, 7);
      const float temporal = __sinf(ct * 50.0f) * m1.a * 0.5f;
      const float chaosn   = m1.b * csd * __sinf(ct * 100.0f);
      const float uburst   = u01(mix3(nid, tu, 107u));
      F2 m2 = nrm2(nid, tu, 8);
      const float burst = (uburst < bp) ? (bc * m2.a * 1.5f) : 0.0f;
      const float noise = lognorm + indiv + temporal + chaosn + burst;

      v = a_mem * (v + 65.0f) - 65.0f + (1.0f - a_mem) * syn + noise - adapt;
      const float v_th = -50.0f + th_b + vth_o + m2.b * tn;

      const float spk = ((v - v_th) >= 0.0f && refr <= 0.0f) ? 1.0f : 0.0f;
      const bool fired = spk > 0.0f;
      v     = fired ? -70.0f : v;
      refr  = fired ? 0.002f : fmaxf(refr - DT, 0.0f);
      adapt = adapt * ADAPT_DEC + astr * spk;
      stdp  = stdp  * STDP_DEC  + 0.01f * spk;

      // single-touch output stream: bypass cache retention (th NT)
      __builtin_nontemporal_store(spk, &obase[(size_t)t * Hn]);
    }

#if HAVE_TDM
    if (tile + 1 < NT) {
      if (wave0) __builtin_amdgcn_s_wait_tensorcnt(0);  // next buffer landed
      __syncthreads();   // publish LDS to all waves; also guards reuse of buf
    }
#endif
  }
}

} // namespace

extern "C" void kernel_launch(void* const* d_in, const int* in_sizes, int n_in,
                              void* d_out, int out_size, void* d_ws, size_t ws_size,
                              hipStream_t stream) {
  (void)in_sizes; (void)n_in; (void)out_size; (void)d_ws; (void)ws_size;

  const float* inp     = (const float*)d_in[0];
  const float* amv     = (const float*)d_in[1];
  const float* asv     = (const float*)d_in[2];
  const float* mnoise  = (const float*)d_in[3];
  const float* snoise  = (const float*)d_in[4];
  const float* tnoise  = (const float*)d_in[5];
  const float* pink    = (const float*)d_in[6];
  const float* jit     = (const float*)d_in[7];
  const float* homeo   = (const float*)d_in[8];
  const float* tbias   = (const float*)d_in[9];
  const float* astr    = (const float*)d_in[10];
  const float* bprob   = (const float*)d_in[11];
  const float* bchaos  = (const float*)d_in[12];
  const float* phase   = (const float*)d_in[13];
  const float* seed    = (const float*)d_in[14];
  const float* vth_off = (const float*)d_in[15];
  const float* nfac    = (const float*)d_in[16];
  float*       out     = (float*)d_out;

  dim3 grid((Bn * Hn) / BLK);   // 64 blocks x 256 threads = one thread per neuron
  dim3 block(BLK);
  hipLaunchKernelGGL(lif_scan_kernel, grid, block, 0, stream,
                     inp, amv, asv, mnoise, snoise, tnoise, pink, jit, homeo,
                     tbias, astr, bprob, bchaos, phase, seed, vth_off, nfac, out);
}